// OCP_15556371546267
// MI455X (gfx1250) — compile-verified
//
#include <hip/hip_runtime.h>
#include <hip/hip_bf16.h>

// Problem constants (from reference): NX=64, NU=32, N=32, BATCH=192, 2 Newton iters
#define NXD 64
#define NUD 32
#define NSTEP 32
#define NBATCH 192
#define NROWS (NSTEP * NBATCH) /* 6144 */

typedef float v2f __attribute__((ext_vector_type(2)));
typedef float v8f __attribute__((ext_vector_type(8)));

// ---------------------------------------------------------------------------
// 16x16 fp32 WMMA tile accumulate: C += A(16xK) * B(Kx16), K in steps of 4.
// Operand layout per CDNA5 ISA (V_WMMA_F32_16X16X4_F32):
//   A 16x4:  v0 <- (m=lr, k=k0+2h), v1 <- (m=lr, k=k0+2h+1)   (h = lane>>4)
//   B 4x16:  v0 <- (k=k0+2h, n=lr), v1 <- (k=k0+2h+1, n=lr)
//   C 16x16: vgpr v holds row (v + 8h), col lr
// ---------------------------------------------------------------------------
template <typename LA, typename LB>
__device__ __forceinline__ v8f wmma_acc(int kdim, LA la, LB lb, v8f acc) {
  const int lane = threadIdx.x & 31;
  const int h2 = (lane >> 4) << 1; // 0 or 2
  const int lr = lane & 15;
  for (int k0 = 0; k0 < kdim; k0 += 4) {
    v2f a, b;
    a.x = la(lr, k0 + h2);
    a.y = la(lr, k0 + h2 + 1);
    b.x = lb(k0 + h2, lr);
    b.y = lb(k0 + h2 + 1, lr);
    acc = __builtin_amdgcn_wmma_f32_16x16x4_f32(false, a, false, b, (short)0,
                                                acc, false, false);
  }
  return acc;
}

template <typename F>
__device__ __forceinline__ void tile_store(const v8f &acc, F f) {
  const int lane = threadIdx.x & 31;
  const int cc = lane & 15;
  const int rbase = (lane >> 4) << 3;
#pragma unroll
  for (int v = 0; v < 8; ++v) f(rbase + v, cc, acc[v]);
}

// ---------------------------------------------------------------------------
// Copy inputs x/u/lmd into mutable workspace (inputs must not be mutated).
// ---------------------------------------------------------------------------
__global__ void ocp_init(const float *__restrict__ x, const float *__restrict__ u,
                         const float *__restrict__ l, float *__restrict__ xw,
                         float *__restrict__ uw, float *__restrict__ lw) {
  const size_t stride = (size_t)gridDim.x * blockDim.x;
  const size_t base = (size_t)blockIdx.x * blockDim.x + threadIdx.x;
  for (size_t i = base; i < (size_t)(NSTEP + 1) * NBATCH * NXD; i += stride) {
    xw[i] = x[i];
    lw[i] = l[i];
  }
  for (size_t i = base; i < (size_t)NSTEP * NBATCH * NUD; i += stride) uw[i] = u[i];
}

// ---------------------------------------------------------------------------
// KKT residuals. Blocks 0..383: 16-row slabs of the (s*B+b) row space, fused
// WMMA GEMMs for xres / lx / lu. Blocks 384..395: x0res (elementwise) + lxN.
// ---------------------------------------------------------------------------
__global__ __launch_bounds__(256) void ocp_kkt(
    const float *__restrict__ x0g, const float *__restrict__ Ag,
    const float *__restrict__ Bg, const float *__restrict__ Qxg,
    const float *__restrict__ Rug, const float *__restrict__ Qfg,
    const float *__restrict__ xw, const float *__restrict__ uw,
    const float *__restrict__ lw, float *__restrict__ x0res,
    float *__restrict__ xres, float *__restrict__ lx, float *__restrict__ lu,
    float *__restrict__ lxN) {
  const int tid = threadIdx.x;
  const int wave = tid >> 5;
  if (blockIdx.x < NROWS / 16) {
    const int r0 = blockIdx.x * 16;
    // tasks: 0..3 xres col-tiles, 4..7 lx col-tiles, 8..9 lu col-tiles
    for (int t = wave; t < 10; t += 8) {
      v8f acc = {};
      if (t < 4) {
        const int n0 = t * 16;
        // xres = x_s A^T + u_s Bm^T - x_{s+1}
        acc = wmma_acc(
            NXD, [&](int r, int k) { return xw[(size_t)(r0 + r) * NXD + k]; },
            [&](int k, int c) { return Ag[(n0 + c) * NXD + k]; }, acc);
        acc = wmma_acc(
            NUD, [&](int r, int k) { return uw[(size_t)(r0 + r) * NUD + k]; },
            [&](int k, int c) { return Bg[(n0 + c) * NUD + k]; }, acc);
        tile_store(acc, [&](int rr, int cc, float val) {
          const int row = r0 + rr;
          const int col = n0 + cc;
          xres[(size_t)row * NXD + col] =
              val - xw[(size_t)(row + NBATCH) * NXD + col];
        });
      } else if (t < 8) {
        const int n0 = (t - 4) * 16;
        // lx = x_s Qx^T + lmd_{s+1} A - lmd_s
        acc = wmma_acc(
            NXD, [&](int r, int k) { return xw[(size_t)(r0 + r) * NXD + k]; },
            [&](int k, int c) { return Qxg[(n0 + c) * NXD + k]; }, acc);
        acc = wmma_acc(
            NXD,
            [&](int r, int k) { return lw[(size_t)(r0 + r + NBATCH) * NXD + k]; },
            [&](int k, int c) { return Ag[k * NXD + (n0 + c)]; }, acc);
        tile_store(acc, [&](int rr, int cc, float val) {
          const int row = r0 + rr;
          const int col = n0 + cc;
          lx[(size_t)row * NXD + col] = val - lw[(size_t)row * NXD + col];
        });
      } else {
        const int n0 = (t - 8) * 16;
        // lu = u_s Ru^T + lmd_{s+1} Bm
        acc = wmma_acc(
            NUD, [&](int r, int k) { return uw[(size_t)(r0 + r) * NUD + k]; },
            [&](int k, int c) { return Rug[(n0 + c) * NUD + k]; }, acc);
        acc = wmma_acc(
            NXD,
            [&](int r, int k) { return lw[(size_t)(r0 + r + NBATCH) * NXD + k]; },
            [&](int k, int c) { return Bg[k * NUD + (n0 + c)]; }, acc);
        tile_store(acc, [&](int rr, int cc, float val) {
          const int row = r0 + rr;
          const int col = n0 + cc;
          lu[(size_t)row * NUD + col] = val;
        });
      }
    }
  } else {
    const int b0 = (blockIdx.x - NROWS / 16) * 16;
    for (int idx = tid; idx < 16 * NXD; idx += 256) {
      const int r = b0 + (idx >> 6);
      const int c = idx & 63;
      x0res[(size_t)r * NXD + c] =
          xw[(size_t)r * NXD + c] - x0g[(size_t)r * NXD + c];
    }
    for (int t = wave; t < 4; t += 8) {
      const int n0 = t * 16;
      v8f acc = {};
      // lxN = x_N Qf^T - lmd_N
      acc = wmma_acc(
          NXD,
          [&](int r, int k) { return xw[(size_t)(NROWS + b0 + r) * NXD + k]; },
          [&](int k, int c) { return Qfg[(n0 + c) * NXD + k]; }, acc);
      tile_store(acc, [&](int rr, int cc, float val) {
        const int row = b0 + rr;
        const int col = n0 + cc;
        lxN[(size_t)row * NXD + col] =
            val - lw[(size_t)(NROWS + row) * NXD + col];
      });
    }
  }
}

// ---------------------------------------------------------------------------
// Backward Riccati recursion: one workgroup (8 waves) per batch element.
// All per-step 64/32-dim matmuls via fp32 WMMA on LDS-resident matrices.
// ---------------------------------------------------------------------------
__global__ __launch_bounds__(256) void ocp_bwd(
    const float *__restrict__ Ag, const float *__restrict__ Bg,
    const float *__restrict__ Qxg, const float *__restrict__ Rug,
    const float *__restrict__ Qfg, const float *__restrict__ xres,
    const float *__restrict__ lxg, const float *__restrict__ lug,
    const float *__restrict__ lxNg, float *__restrict__ Psg,
    float *__restrict__ psg, float *__restrict__ Ksg, float *__restrict__ ksg) {
  __shared__ float sP[NXD * NXD];   // P; later Qxx
  __shared__ float sM1[NXD * NXD];  // A^T P; later Pn
  __shared__ float sPB[NXD * NUD];  // P Bm
  __shared__ float sQxu[NXD * NUD]; // (A^T P) Bm
  __shared__ float sQuu[NUD * 33];  // Quu (padded), Cholesky in place
  __shared__ float sKk[NUD * 65];   // [K | k] (and rhs/solve in place)
  __shared__ float sw[NXD], sqx[NXD], squ[NUD], spv[NXD], sxr[NXD];

  const int tid = threadIdx.x;
  const int wave = tid >> 5;
  const int b = blockIdx.x;

  // P_N = Qf, p_N = lxN ; store to Ps/ps
  {
    const size_t rbN = (size_t)NSTEP * NBATCH + b;
    for (int idx = tid; idx < NXD * NXD; idx += 256) {
      const float q = Qfg[idx];
      sP[idx] = q;
      Psg[(rbN * NXD + (idx >> 6)) * NXD + (idx & 63)] = q;
    }
    if (tid < NXD) {
      const float v0 = lxNg[(size_t)b * NXD + tid];
      spv[tid] = v0;
      psg[rbN * NXD + tid] = v0;
    }
  }
  __syncthreads();

  for (int s = NSTEP - 1; s >= 0; --s) {
    const size_t rb = (size_t)s * NBATCH + b;

    // ---- phase 0: w = P xres + p
    if (tid < NXD) sxr[tid] = xres[rb * NXD + tid];
    __syncthreads();
    if (tid < NXD) {
      float a = spv[tid];
      for (int j = 0; j < NXD; ++j) a += sP[tid * NXD + j] * sxr[j];
      sw[tid] = a;
    }
    __syncthreads();

    // ---- phase 1: M1 = A^T P (16 tiles), PB = P Bm (8 tiles)
    for (int t = wave; t < 24; t += 8) {
      v8f acc = {};
      if (t < 16) {
        const int m0 = (t >> 2) * 16, n0 = (t & 3) * 16;
        acc = wmma_acc(
            NXD, [&](int r, int k) { return Ag[k * NXD + (m0 + r)]; },
            [&](int k, int c) { return sP[k * NXD + (n0 + c)]; }, acc);
        tile_store(acc, [&](int rr, int cc, float val) {
          sM1[(m0 + rr) * NXD + (n0 + cc)] = val;
        });
      } else {
        const int tt = t - 16;
        const int m0 = (tt >> 1) * 16, n0 = (tt & 1) * 16;
        acc = wmma_acc(
            NXD, [&](int r, int k) { return sP[(m0 + r) * NXD + k]; },
            [&](int k, int c) { return Bg[k * NUD + (n0 + c)]; }, acc);
        tile_store(acc, [&](int rr, int cc, float val) {
          sPB[(m0 + rr) * NUD + (n0 + cc)] = val;
        });
      }
    }
    __syncthreads();

    // ---- phase 2: qx/qu (VALU); Qxx = Qx + M1 A -> sP (16 tiles);
    //               Qxu = M1 Bm (8); Quu = Ru + Bm^T PB (4)
    if (tid < NXD) {
      float a = lxg[rb * NXD + tid];
      for (int k = 0; k < NXD; ++k) a += Ag[k * NXD + tid] * sw[k];
      sqx[tid] = a;
    } else if (tid < NXD + NUD) {
      const int uu = tid - NXD;
      float a = lug[rb * NUD + uu];
      for (int k = 0; k < NXD; ++k) a += Bg[k * NUD + uu] * sw[k];
      squ[uu] = a;
    }
    for (int t = wave; t < 28; t += 8) {
      v8f acc = {};
      if (t < 16) {
        const int m0 = (t >> 2) * 16, n0 = (t & 3) * 16;
        acc = wmma_acc(
            NXD, [&](int r, int k) { return sM1[(m0 + r) * NXD + k]; },
            [&](int k, int c) { return Ag[k * NXD + (n0 + c)]; }, acc);
        tile_store(acc, [&](int rr, int cc, float val) {
          sP[(m0 + rr) * NXD + (n0 + cc)] =
              val + Qxg[(m0 + rr) * NXD + (n0 + cc)];
        });
      } else if (t < 24) {
        const int tt = t - 16;
        const int m0 = (tt >> 1) * 16, n0 = (tt & 1) * 16;
        acc = wmma_acc(
            NXD, [&](int r, int k) { return sM1[(m0 + r) * NXD + k]; },
            [&](int k, int c) { return Bg[k * NUD + (n0 + c)]; }, acc);
        tile_store(acc, [&](int rr, int cc, float val) {
          sQxu[(m0 + rr) * NUD + (n0 + cc)] = val;
        });
      } else {
        const int tt = t - 24;
        const int m0 = (tt >> 1) * 16, n0 = (tt & 1) * 16;
        acc = wmma_acc(
            NXD, [&](int r, int k) { return Bg[k * NUD + (m0 + r)]; },
            [&](int k, int c) { return sPB[k * NUD + (n0 + c)]; }, acc);
        tile_store(acc, [&](int rr, int cc, float val) {
          sQuu[(m0 + rr) * 33 + (n0 + cc)] =
              val + Rug[(m0 + rr) * NUD + (n0 + cc)];
        });
      }
    }
    __syncthreads();

    // ---- phase 3: rhs = -[Qxu^T | qu]; Cholesky(Quu); solve; emit K,k
    for (int idx = tid; idx < NUD * 65; idx += 256) {
      const int uu = idx / 65, c = idx % 65;
      sKk[idx] = -((c < NXD) ? sQxu[c * NUD + uu] : squ[uu]);
    }
    __syncthreads();
    for (int k = 0; k < NUD; ++k) {
      if (tid == 0) sQuu[k * 33 + k] = sqrtf(sQuu[k * 33 + k]);
      __syncthreads();
      if (tid > k && tid < NUD) sQuu[tid * 33 + k] /= sQuu[k * 33 + k];
      __syncthreads();
      for (int idx = tid; idx < NUD * NUD; idx += 256) {
        const int i = idx >> 5, j = idx & 31;
        if (i > k && j > k && j <= i)
          sQuu[i * 33 + j] -= sQuu[i * 33 + k] * sQuu[j * 33 + k];
      }
      __syncthreads();
    }
    if (tid < 65) { // 65 independent columns: L y = rhs, then L^T x = y
      for (int r = 0; r < NUD; ++r) {
        float a = sKk[r * 65 + tid];
        for (int t2 = 0; t2 < r; ++t2) a -= sQuu[r * 33 + t2] * sKk[t2 * 65 + tid];
        sKk[r * 65 + tid] = a / sQuu[r * 33 + r];
      }
      for (int r = NUD - 1; r >= 0; --r) {
        float a = sKk[r * 65 + tid];
        for (int t2 = r + 1; t2 < NUD; ++t2)
          a -= sQuu[t2 * 33 + r] * sKk[t2 * 65 + tid];
        sKk[r * 65 + tid] = a / sQuu[r * 33 + r];
      }
    }
    __syncthreads();
    for (int idx = tid; idx < NUD * NXD; idx += 256) {
      const int uu = idx >> 6, xx = idx & 63;
      Ksg[(rb * NUD + uu) * NXD + xx] = sKk[uu * 65 + xx];
    }
    if (tid < NUD) ksg[rb * NUD + tid] = sKk[tid * 65 + 64];

    // ---- phase 4: pn = qx + Qxu k (VALU); Pn = Qxx + Qxu K -> sM1 (16 tiles)
    if (tid < NXD) {
      float a = sqx[tid];
      for (int uu = 0; uu < NUD; ++uu) a += sQxu[tid * NUD + uu] * sKk[uu * 65 + 64];
      spv[tid] = a;
    }
    for (int t = wave; t < 16; t += 8) {
      const int m0 = (t >> 2) * 16, n0 = (t & 3) * 16;
      v8f acc = {};
      acc = wmma_acc(
          NUD, [&](int r, int k) { return sQxu[(m0 + r) * NUD + k]; },
          [&](int k, int c) { return sKk[k * 65 + (n0 + c)]; }, acc);
      tile_store(acc, [&](int rr, int cc, float val) {
        sM1[(m0 + rr) * NXD + (n0 + cc)] =
            val + sP[(m0 + rr) * NXD + (n0 + cc)];
      });
    }
    __syncthreads();

    // ---- phase 5: symmetrize -> sP, store Ps, ps
    for (int idx = tid; idx < NXD * NXD; idx += 256) {
      const int i = idx >> 6, j = idx & 63;
      const float pn = 0.5f * (sM1[i * NXD + j] + sM1[j * NXD + i]);
      sP[idx] = pn;
      Psg[(rb * NXD + i) * NXD + j] = pn;
    }
    if (tid < NXD) psg[rb * NXD + tid] = spv[tid];
    __syncthreads();
  }
}

// ---------------------------------------------------------------------------
// Forward rollout + Newton update (matvecs only): one 64-thread block / batch.
// dx carries forward; dlmd = Ps dx + ps; updates xw/uw/lw; optional u output.
// ---------------------------------------------------------------------------
__global__ __launch_bounds__(64) void ocp_fwd(
    const float *__restrict__ Ag, const float *__restrict__ Bg,
    const float *__restrict__ x0res, const float *__restrict__ xres,
    const float *__restrict__ Psg, const float *__restrict__ psg,
    const float *__restrict__ Ksg, const float *__restrict__ ksg,
    float *__restrict__ xw, float *__restrict__ uw, float *__restrict__ lw,
    float *__restrict__ uout, int write_out) {
  __shared__ float sdx[NXD], sdu[NUD], sdxn[NXD];
  const int tid = threadIdx.x;
  const int b = blockIdx.x;
  sdx[tid] = -x0res[(size_t)b * NXD + tid];
  __syncthreads();
  for (int s = 0; s < NSTEP; ++s) {
    const size_t rb = (size_t)s * NBATCH + b;
    if (tid < NUD) {
      float a = ksg[rb * NUD + tid];
      for (int x = 0; x < NXD; ++x) a += Ksg[(rb * NUD + tid) * NXD + x] * sdx[x];
      sdu[tid] = a;
    }
    __syncthreads();
    {
      float dl = psg[rb * NXD + tid];
      for (int j = 0; j < NXD; ++j) dl += Psg[(rb * NXD + tid) * NXD + j] * sdx[j];
      lw[rb * NXD + tid] += dl;
      xw[rb * NXD + tid] += sdx[tid];
    }
    if (tid < NUD) {
      const float un = uw[rb * NUD + tid] + sdu[tid];
      uw[rb * NUD + tid] = un;
      if (write_out) uout[rb * NUD + tid] = un;
    }
    {
      float a = xres[rb * NXD + tid];
      for (int j = 0; j < NXD; ++j) a += Ag[tid * NXD + j] * sdx[j];
      for (int j = 0; j < NUD; ++j) a += Bg[tid * NUD + j] * sdu[j];
      sdxn[tid] = a;
    }
    __syncthreads();
    sdx[tid] = sdxn[tid];
    __syncthreads();
  }
  {
    const size_t rbN = (size_t)NSTEP * NBATCH + b;
    float dl = psg[rbN * NXD + tid];
    for (int j = 0; j < NXD; ++j) dl += Psg[(rbN * NXD + tid) * NXD + j] * sdx[j];
    lw[rbN * NXD + tid] += dl;
    xw[rbN * NXD + tid] += sdx[tid];
  }
}

// ---------------------------------------------------------------------------
extern "C" void kernel_launch(void *const *d_in, const int *in_sizes, int n_in,
                              void *d_out, int out_size, void *d_ws,
                              size_t ws_size, hipStream_t stream) {
  const float *x0 = (const float *)d_in[0];
  const float *x = (const float *)d_in[1];
  const float *u = (const float *)d_in[2];
  const float *lmd = (const float *)d_in[3];
  const float *A = (const float *)d_in[4];
  const float *Bm = (const float *)d_in[5];
  const float *Qx = (const float *)d_in[6];
  const float *Ru = (const float *)d_in[7];
  const float *Qf = (const float *)d_in[8];

  float *W = (float *)d_ws;
  size_t o = 0;
  float *xw = W + o;  o += (size_t)(NSTEP + 1) * NBATCH * NXD;
  float *uw = W + o;  o += (size_t)NSTEP * NBATCH * NUD;
  float *lw = W + o;  o += (size_t)(NSTEP + 1) * NBATCH * NXD;
  float *x0r = W + o; o += (size_t)NBATCH * NXD;
  float *xre = W + o; o += (size_t)NSTEP * NBATCH * NXD;
  float *lx = W + o;  o += (size_t)NSTEP * NBATCH * NXD;
  float *lu = W + o;  o += (size_t)NSTEP * NBATCH * NUD;
  float *lxN = W + o; o += (size_t)NBATCH * NXD;
  float *Ps = W + o;  o += (size_t)(NSTEP + 1) * NBATCH * NXD * NXD;
  float *ps = W + o;  o += (size_t)(NSTEP + 1) * NBATCH * NXD;
  float *Ks = W + o;  o += (size_t)NSTEP * NBATCH * NUD * NXD;
  float *ks = W + o;  o += (size_t)NSTEP * NBATCH * NUD;

  ocp_init<<<512, 256, 0, stream>>>(x, u, lmd, xw, uw, lw);
  for (int it = 0; it < 2; ++it) {
    ocp_kkt<<<NROWS / 16 + NBATCH / 16, 256, 0, stream>>>(
        x0, A, Bm, Qx, Ru, Qf, xw, uw, lw, x0r, xre, lx, lu, lxN);
    ocp_bwd<<<NBATCH, 256, 0, stream>>>(A, Bm, Qx, Ru, Qf, xre, lx, lu, lxN,
                                        Ps, ps, Ks, ks);
    ocp_fwd<<<NBATCH, 64, 0, stream>>>(A, Bm, x0r, xre, Ps, ps, Ks, ks, xw, uw,
                                       lw, (float *)d_out, it == 1 ? 1 : 0);
  }
  (void)in_sizes; (void)n_in; (void)out_size; (void)ws_size;
}